// PromptGetter_61641370632331
// MI455X (gfx1250) — compile-verified
//
#include <hip/hip_runtime.h>
#include <hip/hip_bf16.h>
#include <math.h>

typedef __attribute__((ext_vector_type(2))) float v2f;
typedef __attribute__((ext_vector_type(8))) float v8f;

#define C_IN    256
#define H_IN    256
#define W_IN    256
#define HW_IN   (H_IN * W_IN)
#define IMG     4096
#define FLATN   (IMG * IMG)
#define NW      64
#define NCELLS_USED 4096          // 64 x 64 occupied grid cells
#define NCELLS      4160          // (n_h + 1) * n_w segments

// ---------------------------------------------------------------------------
// K1: sim[p] = (ref . E[:,p]) / ||E[:,p]||   via v_wmma_f32_16x16x4_f32
// One wave32 handles 16 pixels (N dim). A-matrix row M=0 carries ref; per ISA
// f32 16x4 A layout lane0 holds K=0,1 and lane16 holds K=2,3 of each chunk.
// B-matrix 4x16 chunk of E: lanes 0-15 hold K=0,1 (v0,v1), lanes 16-31 K=2,3.
// Result row M=0 lives in acc[0] of lanes 0-15 (N = lane).
// Sum-of-squares accumulated in VALU from the same loads; halves combined
// with one shfl_xor(16).
// ---------------------------------------------------------------------------
__global__ __launch_bounds__(256) void sim_kernel(const float* __restrict__ emb,
                                                  const float* __restrict__ ref,
                                                  float* __restrict__ sim) {
    __shared__ float refs[C_IN];
    refs[threadIdx.x] = ref[threadIdx.x];
    __syncthreads();

    const int lane   = threadIdx.x & 31;
    const int wave   = blockIdx.x * (blockDim.x >> 5) + (threadIdx.x >> 5);
    const int col    = (wave << 4) + (lane & 15);          // pixel this lane serves
    const int rowOff = (lane < 16) ? 0 : 2;                // K sub-rows this lane loads
    const bool sel   = ((lane & 15) == 0);                 // lanes 0 and 16 carry A row 0

    v8f acc = {0.f, 0.f, 0.f, 0.f, 0.f, 0.f, 0.f, 0.f};
    float ss = 0.f;

    for (int k = 0; k < C_IN; k += 4) {
        float b0 = emb[(size_t)(k + rowOff)     * HW_IN + col];
        float b1 = emb[(size_t)(k + rowOff + 1) * HW_IN + col];
        float r0 = refs[k + rowOff];
        float r1 = refs[k + rowOff + 1];
        v2f av; av.x = sel ? r0 : 0.f; av.y = sel ? r1 : 0.f;
        v2f bv; bv.x = b0;             bv.y = b1;
        acc = __builtin_amdgcn_wmma_f32_16x16x4_f32(
            /*neg_a=*/false, av, /*neg_b=*/false, bv,
            /*c_mod=*/(short)0, acc, /*reuse_a=*/false, /*reuse_b=*/false);
        ss = fmaf(b0, b0, fmaf(b1, b1, ss));
    }

    float ssTot = ss + __shfl_xor(ss, 16, 32);             // combine half-wave partials
    if (lane < 16) {
        sim[col] = acc[0] / sqrtf(ssTot);
    }
}

// ---------------------------------------------------------------------------
// K2: per 64x64-pixel cell of the virtual 4096x4096 bilinear-upsampled mask:
//  - max value > thr with min-flat-index tiebreak  (segment_max + segment_min)
//  - unconditional min value with min-index tiebreak (for bg top_k(-flat,1))
// Bilinear is computed on the fly from the 256KB sim map (L2 resident).
// ---------------------------------------------------------------------------
__global__ __launch_bounds__(256) void cell_kernel(const float* __restrict__ sim,
                                                   const int*   __restrict__ thr_in,
                                                   float* __restrict__ minVal,
                                                   int*   __restrict__ minIdx,
                                                   float* __restrict__ pts) {
    const int b  = blockIdx.x;
    const int cx = b & 63, cy = b >> 6;
    const int tid = threadIdx.x;
    const int t  = thr_in[0];
    const float thr = (t == 0) ? 0.65f : (float)t;

    float bv = -INFINITY; int bi = FLATN;   // best over-threshold (val, idx)
    float mv =  INFINITY; int mi = FLATN;   // global-min candidate (val, idx)

    for (int i = 0; i < 16; ++i) {
        int lid = i * 256 + tid;                       // 0..4095 row-major in cell
        int X = (cx << 6) + (lid & 63);
        int Y = (cy << 6) + (lid >> 6);
        // half-pixel-center bilinear, scale 16, edge clamp
        float u = (X + 0.5f) * 0.0625f - 0.5f;
        float v = (Y + 0.5f) * 0.0625f - 0.5f;
        float fu = floorf(u), fv = floorf(v);
        float du = u - fu,    dv = v - fv;
        int x0 = (int)fu,     y0 = (int)fv;
        int x0c = min(max(x0, 0), W_IN - 1), x1c = min(max(x0 + 1, 0), W_IN - 1);
        int y0c = min(max(y0, 0), H_IN - 1), y1c = min(max(y0 + 1, 0), H_IN - 1);
        float s00 = sim[y0c * W_IN + x0c], s01 = sim[y0c * W_IN + x1c];
        float s10 = sim[y1c * W_IN + x0c], s11 = sim[y1c * W_IN + x1c];
        float top = s00 + (s01 - s00) * du;
        float bot = s10 + (s11 - s10) * du;
        float val = top + (bot - top) * dv;
        int idx = Y * IMG + X;                         // idx increases with i
        if (val > thr && val > bv) { bv = val; bi = idx; }
        if (val < mv)              { mv = val; mi = idx; }
    }

    __shared__ float sBV[256]; __shared__ int sBI[256];
    __shared__ float sMV[256]; __shared__ int sMI[256];
    sBV[tid] = bv; sBI[tid] = bi; sMV[tid] = mv; sMI[tid] = mi;
    __syncthreads();
    for (int s = 128; s > 0; s >>= 1) {
        if (tid < s) {
            float ov = sBV[tid + s]; int oi = sBI[tid + s];
            if (ov > sBV[tid] || (ov == sBV[tid] && oi < sBI[tid])) { sBV[tid] = ov; sBI[tid] = oi; }
            float nv = sMV[tid + s]; int ni = sMI[tid + s];
            if (nv < sMV[tid] || (nv == sMV[tid] && ni < sMI[tid])) { sMV[tid] = nv; sMI[tid] = ni; }
        }
        __syncthreads();
    }
    if (tid == 0) {
        minVal[b] = sMV[0]; minIdx[b] = sMI[0];
        bool occ = sBI[0] < FLATN;
        int idx  = occ ? sBI[0] : 0;
        pts[b * 3 + 0] = occ ? (float)(idx % IMG) : -1.0f;
        pts[b * 3 + 1] = occ ? (float)(idx / IMG) : -1.0f;
        pts[b * 3 + 2] = occ ? sBV[0]             : -1.0f;
    }
}

// K3: segments 4096..4159 can never be occupied -> (-1,-1,-1)
__global__ void fill_tail(float* __restrict__ pts) {
    int i = NCELLS_USED + threadIdx.x;
    if (i < NCELLS) { pts[i*3+0] = -1.0f; pts[i*3+1] = -1.0f; pts[i*3+2] = -1.0f; }
}

// K4: reduce 4096 per-cell minima -> bg point = (x, y) of global min (min-idx ties)
__global__ __launch_bounds__(256) void bg_kernel(const float* __restrict__ minVal,
                                                 const int*   __restrict__ minIdx,
                                                 float* __restrict__ out) {
    int tid = threadIdx.x;
    float mv = INFINITY; int mi = FLATN;
    for (int i = tid; i < NCELLS_USED; i += 256) {
        float v = minVal[i]; int ix = minIdx[i];
        if (v < mv || (v == mv && ix < mi)) { mv = v; mi = ix; }
    }
    __shared__ float sV[256]; __shared__ int sI[256];
    sV[tid] = mv; sI[tid] = mi;
    __syncthreads();
    for (int s = 128; s > 0; s >>= 1) {
        if (tid < s) {
            if (sV[tid+s] < sV[tid] || (sV[tid+s] == sV[tid] && sI[tid+s] < sI[tid])) {
                sV[tid] = sV[tid+s]; sI[tid] = sI[tid+s];
            }
        }
        __syncthreads();
    }
    if (tid == 0) {
        out[NCELLS * 3 + 0] = (float)(sI[0] % IMG);
        out[NCELLS * 3 + 1] = (float)(sI[0] / IMG);
    }
}

// K5: stable descending rank-sort of 4160 triples by score (matches stable
// argsort(-score)): rank = #{keys greater} + #{equal keys at smaller index}.
__global__ __launch_bounds__(256) void sort_kernel(const float* __restrict__ pts,
                                                   float* __restrict__ out) {
    __shared__ float keys[NCELLS];
    for (int i = threadIdx.x; i < NCELLS; i += 256) keys[i] = pts[i*3+2];
    __syncthreads();
    int i = blockIdx.x * 256 + threadIdx.x;
    if (i >= NCELLS) return;
    float ki = keys[i];
    int rank = 0;
    for (int j = 0; j < NCELLS; ++j) {
        float kj = keys[j];
        rank += (kj > ki) || (kj == ki && j < i);
    }
    out[rank*3 + 0] = pts[i*3 + 0];
    out[rank*3 + 1] = pts[i*3 + 1];
    out[rank*3 + 2] = pts[i*3 + 2];
}

extern "C" void kernel_launch(void* const* d_in, const int* in_sizes, int n_in,
                              void* d_out, int out_size, void* d_ws, size_t ws_size,
                              hipStream_t stream) {
    const float* emb = (const float*)d_in[0];   // (1,256,256,256) f32
    const float* ref = (const float*)d_in[1];   // (1,256) f32
    const int*   thr = (const int*)d_in[3];     // threshold scalar
    float* out = (float*)d_out;                 // 4160*3 + 2 = 12482 floats

    // workspace layout (all 4-byte elements)
    float* sim    = (float*)d_ws;               // 65536
    float* minVal = sim + HW_IN;                // 4096
    int*   minIdx = (int*)(minVal + NCELLS_USED);
    float* pts    = (float*)(minIdx + NCELLS_USED);  // 4160*3

    sim_kernel <<<HW_IN / (16 * 8), 256, 0, stream>>>(emb, ref, sim);
    cell_kernel<<<NCELLS_USED,      256, 0, stream>>>(sim, thr, minVal, minIdx, pts);
    fill_tail  <<<1,                 64, 0, stream>>>(pts);
    bg_kernel  <<<1,                256, 0, stream>>>(minVal, minIdx, out);
    sort_kernel<<<(NCELLS + 255)/256, 256, 0, stream>>>(pts, out);
}